// InsertionDeletionSubstitutionGallager_22686017258001
// MI455X (gfx1250) — compile-verified
//
#include <hip/hip_runtime.h>
#include <hip/hip_bf16.h>
#include <stdint.h>

// InsertionDeletionSubstitutionGallager: per-row IDS channel + stream compaction.
// B rows, N=1024 positions, OUT=2048 padded output (pad symbol 4).
// Memory-bound (~256 MB total traffic -> ~11us floor at 23.3 TB/s on MI455X).
// Strategy: 1 wave32 per row; int4 (b128) coalesced loads; per-chunk wave scan;
// BRANCHLESS compaction scatter into a wave-private LDS row buffer (un-kept
// stores steered to a dummy slot via v_cndmask on the address -> no exec-mask
// churn); drain to HBM with CDNA5 GLOBAL_STORE_ASYNC_FROM_LDS_B128 (ASYNCcnt)
// so the output never round-trips through VGPRs.

#define IDS_N        1024
#define IDS_OUT      2048
#define IDS_PAD      4
#define IDS_DUMMY    IDS_OUT        // scratch slot, outside the drained region
#define IDS_STRIDE   (IDS_OUT + 4)  // row stride in LDS (dummy + bank skew)
#define ROWS_PER_BLK 4              // 4 waves * 32 lanes = 128 threads

// ---- gfx1250 feature detection -------------------------------------------
#if defined(__has_builtin)
#  if __has_builtin(__builtin_amdgcn_global_store_async_from_lds_b128)
#    define HAVE_ASYNC_STORE 1
#  else
#    define HAVE_ASYNC_STORE 0
#  endif
#else
#  define HAVE_ASYNC_STORE 0
#endif

// Builtin signature (per hipcc diagnostic): v4i in addrspace(1) dst,
// v4i in addrspace(3) src.
typedef int v4i __attribute__((ext_vector_type(4)));
typedef __attribute__((address_space(1))) v4i* g_v4i_p;   // global int4*
typedef __attribute__((address_space(3))) v4i* l_v4i_p;   // LDS int4*

static __device__ __forceinline__ void wait_dscnt0() {
#if defined(__has_builtin) && __has_builtin(__builtin_amdgcn_s_wait_dscnt)
  __builtin_amdgcn_s_wait_dscnt(0);
#else
  asm volatile("s_wait_dscnt 0" ::: "memory");
#endif
}

static __device__ __forceinline__ void wait_asynccnt0() {
#if defined(__has_builtin) && __has_builtin(__builtin_amdgcn_s_wait_asynccnt)
  __builtin_amdgcn_s_wait_asynccnt(0);
#else
  asm volatile("s_wait_asynccnt 0" ::: "memory");
#endif
}

__global__ __launch_bounds__(32 * ROWS_PER_BLK)
void ids_gallager_kernel(const int* __restrict__ x,
                         const int* __restrict__ mask,
                         const int* __restrict__ ins_rand,
                         const int* __restrict__ sub_off,
                         int* __restrict__ out,
                         int B) {
  __shared__ int s_row[ROWS_PER_BLK][IDS_STRIDE];   // ~32KB + dummy slots

  const int lane = threadIdx.x & 31;             // gfx1250 is wave32
  const int wid  = threadIdx.x >> 5;
  const int row  = blockIdx.x * ROWS_PER_BLK + wid;
  if (row >= B) return;

  int* __restrict__ rlds = &s_row[wid][0];

  const int4* m4p  = (const int4*)(mask     + (size_t)row * IDS_N);
  const int4* x4p  = (const int4*)(x        + (size_t)row * IDS_N);
  const int4* ir4p = (const int4*)(ins_rand + (size_t)row * (2 * IDS_N));
  const int4* so4p = (const int4*)(sub_off  + (size_t)row * (2 * IDS_N));

  // Branchless per-position emit: two unconditional DS stores whose addresses
  // are selected between the compaction cursor and a dummy slot.
  //   ins(0): keep irA, irB   del(1): keep none
  //   sub(2): keep (x+so+1)&3 clean(3): keep x
  auto emit = [&](int m, int xv, int irA, int irB, int so, int& o) {
    const int k0 = (m != 1) ? 1 : 0;            // even slot kept?
    const int k1 = (m == 0) ? 1 : 0;            // odd slot kept? (insertion)
    const int v0 = (m == 0) ? irA
                 : ((m == 2) ? ((xv + so + 1) & 3) : xv);
    const int a0 = k0 ? o : IDS_DUMMY;          // v_cndmask, no branch
    const int a1 = k1 ? (o + 1) : IDS_DUMMY;
    rlds[a0] = v0;
    rlds[a1] = irB;
    o += k0 + k1;
  };

  int running = 0;                     // row-level carry of the scan
  #pragma unroll
  for (int c = 0; c < IDS_N / 128; ++c) {        // 8 chunks of 128 positions
    const int g = c * 32 + lane;                 // int4 group index (4 pos/lane)
    const int4 m4   = m4p[g];
    const int4 x4   = x4p[g];
    const int4 irlo = ir4p[2 * g];               // slots 2p..2p+3
    const int4 irhi = ir4p[2 * g + 1];           // slots 2p+4..2p+7
    const int4 solo = so4p[2 * g];               // even slots: .x, .z
    const int4 sohi = so4p[2 * g + 1];

    // per-lane kept count (0..8), branchless
    const int L =
        ((m4.x != 1) ? 1 : 0) + ((m4.x == 0) ? 1 : 0) +
        ((m4.y != 1) ? 1 : 0) + ((m4.y == 0) ? 1 : 0) +
        ((m4.z != 1) ? 1 : 0) + ((m4.z == 0) ? 1 : 0) +
        ((m4.w != 1) ? 1 : 0) + ((m4.w == 0) ? 1 : 0);

    // wave32 inclusive scan of per-lane counts
    int incl = L;
    #pragma unroll
    for (int d = 1; d < 32; d <<= 1) {
      int t = __shfl_up(incl, d, 32);
      if (lane >= d) incl += t;
    }

    int o = running + (incl - L);                // exclusive offset for this lane
    emit(m4.x, x4.x, irlo.x, irlo.y, solo.x, o);
    emit(m4.y, x4.y, irlo.z, irlo.w, solo.z, o);
    emit(m4.z, x4.z, irhi.x, irhi.y, sohi.x, o);
    emit(m4.w, x4.w, irhi.z, irhi.w, sohi.z, o);

    running += __shfl(incl, 31, 32);             // broadcast chunk total
  }

  // Pad the tail of the compacted row. running <= 2*N = 2048.
  for (int i = running + lane; i < IDS_OUT; i += 32) {
    rlds[i] = IDS_PAD;
  }

  // Make all LDS writes visible to the async DMA engine, then drain the
  // 8KB row buffer to HBM with coalesced 128-bit LDS->global async stores.
  asm volatile("" ::: "memory");
  wait_dscnt0();

  int* __restrict__ out_row = out + (size_t)row * IDS_OUT;
#if HAVE_ASYNC_STORE
  #pragma unroll
  for (int it = 0; it < IDS_OUT / 128; ++it) {   // 16 x (32 lanes x 16B) = 8KB
    const int idx = it * 128 + lane * 4;
    __builtin_amdgcn_global_store_async_from_lds_b128(
        (g_v4i_p)(out_row + idx), (l_v4i_p)(rlds + idx), 0, 0);
  }
  wait_asynccnt0();
#else
  // Fallback: b128 loads from LDS + b128 global stores (still fully coalesced).
  #pragma unroll
  for (int it = 0; it < IDS_OUT / 128; ++it) {
    const int idx = it * 128 + lane * 4;
    *(int4*)(out_row + idx) = *(const int4*)(rlds + idx);
  }
#endif
}

extern "C" void kernel_launch(void* const* d_in, const int* in_sizes, int n_in,
                              void* d_out, int out_size, void* d_ws, size_t ws_size,
                              hipStream_t stream) {
  (void)d_ws; (void)ws_size; (void)out_size;
  if (n_in < 4) return;

  const int* x        = (const int*)d_in[0];   // [B, N, 1] int32
  const int* mask     = (const int*)d_in[1];   // [B, N]    int32
  const int* ins_rand = (const int*)d_in[2];   // [B, 2N]   int32
  const int* sub_off  = (const int*)d_in[3];   // [B, 2N]   int32
  int*       out      = (int*)d_out;           // [B, 2048, 1] int32

  const int B = in_sizes[0] / IDS_N;

  dim3 block(32 * ROWS_PER_BLK);
  dim3 grid((B + ROWS_PER_BLK - 1) / ROWS_PER_BLK);
  ids_gallager_kernel<<<grid, block, 0, stream>>>(x, mask, ins_rand, sub_off, out, B);
}